// HingeCalibratedRanking_2869038153762
// MI455X (gfx1250) — compile-verified
//
#include <hip/hip_runtime.h>
#include <hip/hip_bf16.h>

typedef float v2f __attribute__((ext_vector_type(2)));
typedef float v8f __attribute__((ext_vector_type(8)));

#define NROWS 1024
#define LCOLS 512
#define TILES 32          // 512 / 16
#define BLOCK 256         // 8 waves of 32
#define FRAGN (TILES * 32)

__device__ __forceinline__ float block_reduce_sum(float v, float* red, int tid) {
    red[tid] = v;
    __syncthreads();
    #pragma unroll
    for (int s = BLOCK / 2; s > 0; s >>= 1) {
        if (tid < s) red[tid] += red[tid + s];
        __syncthreads();
    }
    float r = red[0];
    __syncthreads();
    return r;
}

__global__ __launch_bounds__(BLOCK)
void hinge_row_kernel(const float* __restrict__ outputs,
                      const int*   __restrict__ targets,
                      float*       __restrict__ row_out) {
    __shared__ float lds_n[LCOLS];    // negatives (sentinel -1e30 where positive)
    __shared__ float lds_p[LCOLS];    // positives (sentinel +1e30 where negative)
    __shared__ v2f   nfrag[FRAGN];    // A fragment {K0=n_j, K1=1}; lanes 16-31 -> {0,0}
    __shared__ v2f   bfrag[FRAGN];    // B fragment {K0=1, K1=1-p_k}; lanes 16-31 -> {0,0}
    __shared__ float red[BLOCK];

    const int b   = blockIdx.x;
    const int tid = threadIdx.x;
    const float* orow = outputs + (size_t)b * LCOLS;
    const int*   trow = targets + (size_t)b * LCOLS;

    // ---- phase 1a: sentinel arrays + calibration partial sums ----
    float pos_sum = 0.0f, neg_sum = 0.0f, np = 0.0f, nn = 0.0f;
    #pragma unroll
    for (int i = 0; i < LCOLS / BLOCK; ++i) {
        const int l = tid + i * BLOCK;
        const float o = orow[l];
        const bool isPos = (trow[l] != 0);
        lds_n[l] = isPos ? -1.0e30f : o;
        lds_p[l] = isPos ? o : 1.0e30f;
        if (isPos) { pos_sum += fmaxf(1.0f - o, 0.0f); np += 1.0f; }
        else       { neg_sum += fmaxf(1.0f + o, 0.0f); nn += 1.0f; }
    }
    __syncthreads();

    // ---- phase 1b: prebuild complete 64-bit WMMA fragments ----
    // One ds_load_b64 per fragment in the hot loop; no per-WMMA register moves.
    #pragma unroll
    for (int i = 0; i < FRAGN / BLOCK; ++i) {
        const int e  = tid + i * BLOCK;
        const int ln = e & 31;
        const int tt = e >> 5;
        const int col = tt * 16 + (ln & 15);
        const float vn = lds_n[col];          // unconditional loads -> no exec branches
        const float vp = lds_p[col];
        const bool  lo2 = (ln < 16);
        v2f an, bk;
        an.x = lo2 ? vn : 0.0f;               // A[:,K=0] = n_j    (K=2 half zeroed)
        an.y = lo2 ? 1.0f : 0.0f;             // A[:,K=1] = 1      (K=3 half zeroed)
        bk.x = an.y;                          // B[K=0,:] = 1
        bk.y = lo2 ? (1.0f - vp) : 0.0f;      // B[K=1,:] = 1-p_k
        nfrag[e] = an;
        bfrag[e] = bk;
    }
    __syncthreads();

    // ---- phase 2: pairwise hinge via V_WMMA_F32_16X16X4_F32 ----
    // diff[j,k] = 1 + n_j - p_k = [n_j, 1] x [1 ; 1-p_k]  (rank-2)
    const int wave = tid >> 5;
    const int lane = tid & 31;

    v8f accv = {};
    for (int jt = wave; jt < TILES; jt += 8) {
        const v2f a = nfrag[jt * 32 + lane];          // one ds_load_b64 per jt
        #pragma unroll 4
        for (int kt = 0; kt < TILES; ++kt) {
            const v2f bb = bfrag[kt * 32 + lane];     // one ds_load_b64 per tile
            v8f c = {};
            c = __builtin_amdgcn_wmma_f32_16x16x4_f32(
                    /*neg_a=*/false, a, /*neg_b=*/false, bb,
                    /*c_mod=*/(short)0, c, /*reuse_a=*/false, /*reuse_b=*/false);
            #pragma unroll
            for (int i = 0; i < 8; ++i) accv[i] += fmaxf(c[i], 0.0f);
        }
    }
    float acc = 0.0f;
    #pragma unroll
    for (int i = 0; i < 8; ++i) acc += accv[i];

    // ---- phase 3: block reductions + per-row result ----
    const float pair    = block_reduce_sum(acc,     red, tid);
    const float psum    = block_reduce_sum(pos_sum, red, tid);
    const float nsum    = block_reduce_sum(neg_sum, red, tid);
    const float np_tot  = block_reduce_sum(np,      red, tid);
    const float nn_tot  = block_reduce_sum(nn,      red, tid);

    if (tid == 0) {
        const float pos_calib = (np_tot > 0.0f) ? (psum / np_tot) : 0.0f;
        const float neg_calib = (nn_tot > 0.0f) ? (nsum / nn_tot) : 0.0f;
        const float denom = np_tot * nn_tot;
        float hinge;
        if (denom > 0.0f)       hinge = pair / denom;
        else if (np_tot > 0.0f) hinge = pos_calib;
        else if (nn_tot > 0.0f) hinge = neg_calib;
        else                    hinge = 1.0f;
        row_out[b] = hinge + neg_calib + pos_calib;
    }
}

__global__ __launch_bounds__(BLOCK)
void final_mean_kernel(const float* __restrict__ row_out, float* __restrict__ out) {
    __shared__ float red[BLOCK];
    const int tid = threadIdx.x;
    float s = 0.0f;
    #pragma unroll
    for (int i = tid; i < NROWS; i += BLOCK) s += row_out[i];
    red[tid] = s;
    __syncthreads();
    #pragma unroll
    for (int st = BLOCK / 2; st > 0; st >>= 1) {
        if (tid < st) red[tid] += red[tid + st];
        __syncthreads();
    }
    if (tid == 0) out[0] = red[0] * (1.0f / (float)NROWS);
}

extern "C" void kernel_launch(void* const* d_in, const int* in_sizes, int n_in,
                              void* d_out, int out_size, void* d_ws, size_t ws_size,
                              hipStream_t stream) {
    const float* outputs = (const float*)d_in[0];
    const int*   targets = (const int*)d_in[1];
    float* row_out = (float*)d_ws;          // 1024 floats of scratch
    float* out     = (float*)d_out;         // scalar f32

    hinge_row_kernel<<<NROWS, BLOCK, 0, stream>>>(outputs, targets, row_out);
    final_mean_kernel<<<1, BLOCK, 0, stream>>>(row_out, out);
}